// OnnxAttLoc_22917945491700
// MI455X (gfx1250) — compile-verified
//
#include <hip/hip_runtime.h>
#include <math.h>

// ---- problem constants (match reference) ----
#define T_LEN   50000
#define ATT     512
#define EPROJ   512
#define DUNITS  1024
#define CH      10
#define KSZ     201
#define FILTS   100
#define SCALING 2.0f

#define CONV_STRIDE 16          // conv row padded to 16 floats (c=10..15 zero)
#define T_TILES (T_LEN / 16)    // 3125, exact

// workspace layout (in floats)
#define WS_DEC   0
#define WS_CONV  (WS_DEC + 512)
#define WS_E     (WS_CONV + T_LEN * CONV_STRIDE)
#define WS_PART  (WS_E + T_LEN)
#define NB_CTX   256

// transposed w_att in LDS: 12 rows (channels, padded), row stride 520 floats
// (row delta between lane-halves = 2*520 = 1040 ; 1040 % 64 = 16 -> no bank conflict)
#define BSTRIDE 520

typedef __attribute__((ext_vector_type(2))) float v2f;
typedef __attribute__((ext_vector_type(8))) float v8f;

#if __has_builtin(__builtin_amdgcn_tanhf)
static __device__ __forceinline__ float fast_tanh(float x) { return __builtin_amdgcn_tanhf(x); }
#elif __has_builtin(__builtin_amdgcn_tanh_f32)
static __device__ __forceinline__ float fast_tanh(float x) { return __builtin_amdgcn_tanh_f32(x); }
#else
static __device__ __forceinline__ float fast_tanh(float x) { return tanhf(x); }
#endif

// ---------------------------------------------------------------------------
// Kernel 1: dec_proj[a] = sum_d w_dec[a,d] * dec_z[d]  via V_WMMA_F32_16X16X4_F32
// ---------------------------------------------------------------------------
__global__ void dec_proj_wmma_kernel(const float* __restrict__ w_dec,
                                     const float* __restrict__ dec_z,
                                     float* __restrict__ ws_dec) {
    const int lane = threadIdx.x & 31;
    const int gwave = blockIdx.x * (blockDim.x >> 5) + (threadIdx.x >> 5);
    const int m0 = gwave * 16;              // 32 tiles cover ATT=512

    const int row   = m0 + (lane & 15);
    const int khalf = (lane >> 4) << 1;     // 0 for lanes 0-15, 2 for 16-31
    const bool bcol = ((lane & 15) == 0);   // only lanes 0/16 carry B column 0

    v8f acc = {};
    for (int k0 = 0; k0 < DUNITS; k0 += 4) {
        v2f a, b;
        const float* wrow = w_dec + (size_t)row * DUNITS + k0 + khalf;
        a.x = wrow[0];
        a.y = wrow[1];
        float d0 = dec_z[k0 + khalf + 0];
        float d1 = dec_z[k0 + khalf + 1];
        b.x = bcol ? d0 : 0.0f;
        b.y = bcol ? d1 : 0.0f;
        acc = __builtin_amdgcn_wmma_f32_16x16x4_f32(false, a, false, b,
                                                    (short)0, acc, false, false);
    }
    if (lane == 0) {
        #pragma unroll
        for (int v = 0; v < 8; ++v) ws_dec[m0 + v] = acc[v];
    } else if (lane == 16) {
        #pragma unroll
        for (int v = 0; v < 8; ++v) ws_dec[m0 + 8 + v] = acc[v];
    }
}

// ---------------------------------------------------------------------------
// Kernel 2: 1-D conv (CH channels, padding FILTS). 4 t's per thread so each
// LDS-broadcast tap feeds 4 FMAs. Row padded to 16; c=10..15 zeroed so the
// K=12-padded WMMA A-fragments read zeros.
// ---------------------------------------------------------------------------
__global__ void conv_kernel(const float* __restrict__ att_prev,
                            const float* __restrict__ conv_w,
                            float* __restrict__ ws_conv) {
    __shared__ float win[1024 + 2 * FILTS];   // 1224
    __shared__ float scw[CH * KSZ];           // 2010

    const int tid = threadIdx.x;
    const int t0  = blockIdx.x * 1024;

    for (int i = tid; i < 1024 + 2 * FILTS; i += 256) {
        int src = t0 - FILTS + i;
        win[i] = (src >= 0 && src < T_LEN) ? att_prev[src] : 0.0f;
    }
    for (int i = tid; i < CH * KSZ; i += 256) scw[i] = conv_w[i];
    __syncthreads();

    float acc[CH][4];
    #pragma unroll
    for (int c = 0; c < CH; ++c)
        #pragma unroll
        for (int i = 0; i < 4; ++i) acc[c][i] = 0.0f;

    const int base = tid * 4;
    for (int k = 0; k < KSZ; ++k) {
        float w0 = win[base + k + 0];
        float w1 = win[base + k + 1];
        float w2 = win[base + k + 2];
        float w3 = win[base + k + 3];
        #pragma unroll
        for (int c = 0; c < CH; ++c) {
            float cv = scw[c * KSZ + k];
            acc[c][0] = fmaf(cv, w0, acc[c][0]);
            acc[c][1] = fmaf(cv, w1, acc[c][1]);
            acc[c][2] = fmaf(cv, w2, acc[c][2]);
            acc[c][3] = fmaf(cv, w3, acc[c][3]);
        }
    }

    #pragma unroll
    for (int i = 0; i < 4; ++i) {
        int t = t0 + base + i;
        if (t < T_LEN) {
            float* dst = ws_conv + (size_t)t * CONV_STRIDE;
            #pragma unroll
            for (int c = 0; c < CH; ++c) dst[c] = acc[c][i];
            #pragma unroll
            for (int c = CH; c < CONV_STRIDE; ++c) dst[c] = 0.0f;
        }
    }
}

// ---------------------------------------------------------------------------
// Kernel 3: scores via WMMA. One wave per 16-t tile.
//   proj(16t x 512a) = conv(16x12,zero-padded) x w_attT(12x512)  [3 chained
//   v_wmma_f32_16x16x4_f32 per 16x16 N-tile, C preloaded with pre + dec]
//   e[t] = sum_a tanh(proj)*gvec[a] + b + mask[t]
// C/D layout: VGPR r <-> rows (t0+r | t0+r+8) on lane halves, N = lane&15.
// ---------------------------------------------------------------------------
__global__ void scores_wmma_kernel(const float* __restrict__ pre,
                                   const float* __restrict__ mask,
                                   const float* __restrict__ w_att,
                                   const float* __restrict__ w_gvec,
                                   const float* __restrict__ b_gvec,
                                   const float* __restrict__ ws_dec,
                                   const float* __restrict__ ws_conv,
                                   float* __restrict__ ws_e) {
    __shared__ float satt[12 * BSTRIDE];   // transposed w_att, 12 rows, padded
    __shared__ float sdec[ATT];
    __shared__ float sgv[ATT];

    const int tid  = threadIdx.x;
    const int lane = tid & 31;

    // build transposed + zero-padded w_att in LDS: satt[c*BSTRIDE + a]
    for (int i = tid; i < 12 * ATT; i += 256) {
        int c = i >> 9;           // /512
        int a = i & 511;
        satt[c * BSTRIDE + a] = (c < CH) ? w_att[a * CH + c] : 0.0f;
    }
    for (int i = tid; i < ATT; i += 256) {
        sdec[i] = ws_dec[i];
        sgv[i]  = w_gvec[i];
    }
    __syncthreads();

    const float bias = b_gvec[0];
    const int kh     = (lane >> 4) << 1;       // 0 | 2 (K-half per lane group)
    const int col    = lane & 15;              // N (and M for A) index
    const int rowoff = (lane >> 4) << 3;       // 0 | 8 (C/D row offset)

    const int nwaves = gridDim.x * (blockDim.x >> 5);
    const int gwave  = blockIdx.x * (blockDim.x >> 5) + (tid >> 5);

    for (int tile = gwave; tile < T_TILES; tile += nwaves) {
        const int t0 = tile * 16;

        // A fragments: conv rows, K = {0..3},{4..7},{8..11} (10,11 are zero)
        const float* crow = ws_conv + (size_t)(t0 + col) * CONV_STRIDE + kh;
        v2f A0, A1, A2;
        A0.x = crow[0]; A0.y = crow[1];
        A1.x = crow[4]; A1.y = crow[5];
        A2.x = crow[8]; A2.y = crow[9];

        float accr[8];
        #pragma unroll
        for (int r = 0; r < 8; ++r) accr[r] = 0.0f;

        for (int nt = 0; nt < 32; ++nt) {
            const int a0 = nt * 16;

            // B fragments from LDS (row k on lanes0-15, row k+2 on lanes16-31)
            const float* bp = satt + (size_t)kh * BSTRIDE + a0 + col;
            v2f B0, B1, B2;
            B0.x = bp[0 * BSTRIDE];  B0.y = bp[1 * BSTRIDE];
            B1.x = bp[4 * BSTRIDE];  B1.y = bp[5 * BSTRIDE];
            B2.x = bp[8 * BSTRIDE];  B2.y = bp[9 * BSTRIDE];

            const float dec_l = sdec[a0 + col];
            const float gv_l  = sgv[a0 + col];

            // C = pre + dec (the WMMA accumulate adds the projection)
            const float* prow = pre + (size_t)(t0 + rowoff) * ATT + a0 + col;
            v8f C;
            #pragma unroll
            for (int r = 0; r < 8; ++r) C[r] = prow[(size_t)r * ATT] + dec_l;

            v8f D;
            D = __builtin_amdgcn_wmma_f32_16x16x4_f32(false, A0, false, B0,
                                                      (short)0, C, false, false);
            D = __builtin_amdgcn_wmma_f32_16x16x4_f32(false, A1, false, B1,
                                                      (short)0, D, false, false);
            D = __builtin_amdgcn_wmma_f32_16x16x4_f32(false, A2, false, B2,
                                                      (short)0, D, false, false);

            #pragma unroll
            for (int r = 0; r < 8; ++r)
                accr[r] = fmaf(fast_tanh(D[r]), gv_l, accr[r]);
        }

        // reduce over N within each 16-lane half
        #pragma unroll
        for (int r = 0; r < 8; ++r) {
            float v = accr[r];
            v += __shfl_xor(v, 1, 32);
            v += __shfl_xor(v, 2, 32);
            v += __shfl_xor(v, 4, 32);
            v += __shfl_xor(v, 8, 32);
            accr[r] = v;
        }
        if (lane == 0) {
            #pragma unroll
            for (int r = 0; r < 8; ++r)
                ws_e[t0 + r] = accr[r] + bias + mask[t0 + r];
        } else if (lane == 16) {
            #pragma unroll
            for (int r = 0; r < 8; ++r)
                ws_e[t0 + 8 + r] = accr[r] + bias + mask[t0 + 8 + r];
        }
    }
}

// ---------------------------------------------------------------------------
// Kernel 4: softmax over T of SCALING*e ; writes w to d_out+512.
// Single block, fixed-order reductions -> bitwise deterministic.
// ---------------------------------------------------------------------------
__global__ void softmax_kernel(const float* __restrict__ ws_e,
                               float* __restrict__ w_out) {
    __shared__ float red[1024];
    const int tid = threadIdx.x;

    float m = -INFINITY;
    for (int t = tid; t < T_LEN; t += 1024) m = fmaxf(m, SCALING * ws_e[t]);
    red[tid] = m;
    __syncthreads();
    for (int s = 512; s > 0; s >>= 1) {
        if (tid < s) red[tid] = fmaxf(red[tid], red[tid + s]);
        __syncthreads();
    }
    m = red[0];
    __syncthreads();

    float sum = 0.0f;
    for (int t = tid; t < T_LEN; t += 1024) sum += expf(SCALING * ws_e[t] - m);
    red[tid] = sum;
    __syncthreads();
    for (int s = 512; s > 0; s >>= 1) {
        if (tid < s) red[tid] += red[tid + s];
        __syncthreads();
    }
    const float inv = 1.0f / red[0];

    for (int t = tid; t < T_LEN; t += 1024)
        w_out[t] = expf(SCALING * ws_e[t] - m) * inv;
}

// ---------------------------------------------------------------------------
// Kernel 5: per-block partial of c[j] = sum_t enc_h[t,j]*w[t].
// ---------------------------------------------------------------------------
__global__ void context_partial_kernel(const float* __restrict__ enc_h,
                                       const float* __restrict__ w,
                                       float* __restrict__ ws_part) {
    const int tid = threadIdx.x;
    const int per = (T_LEN + NB_CTX - 1) / NB_CTX;
    const int t0  = blockIdx.x * per;
    const int t1  = min(T_LEN, t0 + per);

    float a0 = 0.0f, a1 = 0.0f;
    for (int t = t0; t < t1; ++t) {
        float wt = w[t];
        const float* row = enc_h + (size_t)t * EPROJ;
        // prefetch a few rows ahead (lowers to global_prefetch_b8)
        __builtin_prefetch(row + 8 * EPROJ, 0, 0);
        a0 = fmaf(row[tid],       wt, a0);
        a1 = fmaf(row[tid + 256], wt, a1);
    }
    ws_part[(size_t)blockIdx.x * EPROJ + tid]       = a0;
    ws_part[(size_t)blockIdx.x * EPROJ + tid + 256] = a1;
}

// Kernel 6: fixed-order reduce of partials -> c
__global__ void context_reduce_kernel(const float* __restrict__ ws_part,
                                      float* __restrict__ c_out) {
    const int j = threadIdx.x;   // 512 threads
    float s = 0.0f;
    for (int b = 0; b < NB_CTX; ++b) s += ws_part[(size_t)b * EPROJ + j];
    c_out[j] = s;
}

// ---------------------------------------------------------------------------
extern "C" void kernel_launch(void* const* d_in, const int* in_sizes, int n_in,
                              void* d_out, int out_size, void* d_ws, size_t ws_size,
                              hipStream_t stream) {
    const float* dec_z    = (const float*)d_in[0];
    const float* att_prev = (const float*)d_in[1];
    const float* pre      = (const float*)d_in[2];
    const float* enc_h    = (const float*)d_in[3];
    const float* mask     = (const float*)d_in[4];
    const float* conv_w   = (const float*)d_in[5];
    const float* w_att    = (const float*)d_in[6];
    const float* w_dec    = (const float*)d_in[7];
    const float* w_gvec   = (const float*)d_in[8];
    const float* b_gvec   = (const float*)d_in[9];

    float* out = (float*)d_out;            // [0,512) = c ; [512, 512+T) = w
    float* ws  = (float*)d_ws;
    float* ws_dec  = ws + WS_DEC;
    float* ws_conv = ws + WS_CONV;
    float* ws_e    = ws + WS_E;
    float* ws_part = ws + WS_PART;

    dec_proj_wmma_kernel<<<4, 256, 0, stream>>>(w_dec, dec_z, ws_dec);

    conv_kernel<<<(T_LEN + 1023) / 1024, 256, 0, stream>>>(att_prev, conv_w, ws_conv);

    scores_wmma_kernel<<<400, 256, 0, stream>>>(pre, mask, w_att, w_gvec, b_gvec,
                                                ws_dec, ws_conv, ws_e);

    softmax_kernel<<<1, 1024, 0, stream>>>(ws_e, out + EPROJ);

    context_partial_kernel<<<NB_CTX, 256, 0, stream>>>(enc_h, out + EPROJ, ws_part);
    context_reduce_kernel<<<1, 512, 0, stream>>>(ws_part, out);
}